// NCE_21208548508487
// MI455X (gfx1250) — compile-verified
//
#include <hip/hip_runtime.h>
#include <hip/hip_bf16.h>

// ---------------------------------------------------------------------------
// NCE loss on MI455X (gfx1250).
//
// Roofline: column gathers from embed (128 x 1e6, row stride 4MB) touch one
// 128B line per scalar -> ~512MB of HBM traffic; at 23.3 TB/s this is a
// ~23us memory-bound kernel. FLOPs (~10M) are free, so we spend them on
// V_WMMA_F32_16X16X4_F32: one wave computes 16 paired dot products as the
// diagonal of a 16x16 f32 WMMA accumulation over 32 K=4 chunks (full f32
// precision, matching the f32 reference). All reductions are fixed-order
// (LDS trees + a single final block) -> bitwise deterministic, no atomics.
// ---------------------------------------------------------------------------

typedef float v2f __attribute__((ext_vector_type(2)));
typedef float v8f __attribute__((ext_vector_type(8)));

#define VOCAB_N   1000000
#define EMBED_N   128
#define BATCH_N   16384
#define NOISE_N   16

#define K1_BLOCKS 128          // 128 pairs per block (8 waves * 16)
#define K2_BLOCKS 1024         // 262144 noise samples / 256
#define N_PART    (K1_BLOCKS + K2_BLOCKS)

__device__ __forceinline__ float log_sigmoid_f(float x) {
    // log(sigmoid(x)) = min(x,0) - log1p(exp(-|x|))   (numerically stable)
    return fminf(x, 0.0f) - log1pf(__expf(-fabsf(x)));
}

__device__ __forceinline__ float block_reduce_256(float v, float* sm) {
    const int t = threadIdx.x;
    sm[t] = v;
    __syncthreads();
    #pragma unroll
    for (int s = 128; s > 0; s >>= 1) {
        if (t < s) sm[t] += sm[t + s];
        __syncthreads();
    }
    return sm[0];
}

// Kernel 1: gathers + WMMA dot products + positive loss + L2 penalty.
__global__ __launch_bounds__(256) void nce_dot_kernel(
    const float* __restrict__ embed,   // (E, VOCAB) row-major
    const float* __restrict__ bias,    // (VOCAB, 1)
    const float* __restrict__ freq,    // (VOCAB,)
    const int*   __restrict__ targets, // (B,)
    const int*   __restrict__ contexts,// (B,)
    float*       __restrict__ s_out,   // (B,) workspace
    float*       __restrict__ partials)// (K1_BLOCKS,)
{
    __shared__ float sm[256];

    const int lane  = threadIdx.x & 31;
    const int wave  = threadIdx.x >> 5;
    const int m16   = lane & 15;                       // matrix row/col this lane feeds
    const int elem  = blockIdx.x * 128 + wave * 16 + m16;
    const int tgt   = targets[elem];
    const int ctx   = contexts[elem];
    const int ehalf = (lane >> 4) << 1;                // lanes 0-15: K=0,1  lanes 16-31: K=2,3

    const float* qcol = embed + tgt;                   // column, stride VOCAB
    const float* rcol = embed + ctx;

    v8f   c     = {};                                  // 16x16 f32 accumulator
    float sumsq = 0.0f;

    #pragma unroll 4
    for (int e0 = 0; e0 < EMBED_N; e0 += 4) {
        const size_t off0 = (size_t)(e0 + ehalf) * (size_t)VOCAB_N;
        const size_t off1 = off0 + (size_t)VOCAB_N;

        if (e0 + 4 < EMBED_N) {                        // uniform branch (s_branch)
            __builtin_prefetch(qcol + off0 + 4ull * VOCAB_N, 0, 0);
            __builtin_prefetch(rcol + off0 + 4ull * VOCAB_N, 0, 0);
        }

        v2f a, b;
        a.x = qcol[off0];  a.y = qcol[off1];           // A: 16x4 chunk of Q
        b.x = rcol[off0];  b.y = rcol[off1];           // B: 4x16 chunk of R (mirrored layout)

        sumsq += a.x * a.x + a.y * a.y + b.x * b.x + b.y * b.y;

        // D = A x B + C ; diagonal accumulates q_m . r_m
        c = __builtin_amdgcn_wmma_f32_16x16x4_f32(
                /*neg_a=*/false, a, /*neg_b=*/false, b,
                /*c_mod=*/(short)0, c, /*reuse_a=*/false, /*reuse_b=*/false);
    }

    // Diagonal extraction. C/D layout: lanes 0-15 hold M=VGPR (N=lane),
    // lanes 16-31 hold M=VGPR+8 (N=lane-16). Diag (m,m):
    //   m<8  -> lane m,      VGPR m
    //   m>=8 -> lane m+16,   VGPR m-8
    const int  idx    = lane & 7;
    const bool active = (lane < 8) || (lane >= 24);
    float d = c[0];
    if (idx == 1) d = c[1];
    if (idx == 2) d = c[2];
    if (idx == 3) d = c[3];
    if (idx == 4) d = c[4];
    if (idx == 5) d = c[5];
    if (idx == 6) d = c[6];
    if (idx == 7) d = c[7];

    // per-thread contribution:  10/(E*B) * sumsq  (+ -pos/B on diag lanes)
    const float PEN_SCALE = 10.0f / ((float)EMBED_N * (float)BATCH_N);
    float contrib = PEN_SCALE * sumsq;

    if (active) {
        const float s = (d + bias[tgt]) * (1.0f / (float)EMBED_N);
        s_out[elem] = s;
        const float x  = s - __logf((float)NOISE_N * freq[tgt]);
        contrib += -log_sigmoid_f(x) * (1.0f / (float)BATCH_N);
    }

    const float blocksum = block_reduce_256(contrib, sm);
    if (threadIdx.x == 0) partials[blockIdx.x] = blocksum;
}

// Kernel 2: noise loss.  k -> s[k mod B], freq[noises[k]].
__global__ __launch_bounds__(256) void nce_noise_kernel(
    const float* __restrict__ freq,
    const int*   __restrict__ noises,  // (B*NOISE_N,)
    const float* __restrict__ s,       // (B,) from kernel 1
    float*       __restrict__ partials)// (K2_BLOCKS,) at offset K1_BLOCKS
{
    __shared__ float sm[256];
    const int k = blockIdx.x * 256 + threadIdx.x;      // k < B*NOISE_N
    const int b = k & (BATCH_N - 1);
    const int n = noises[k];
    const float x = -(s[b] - __logf((float)NOISE_N * freq[n]));
    const float contrib = -log_sigmoid_f(x) * (1.0f / (float)BATCH_N);

    const float blocksum = block_reduce_256(contrib, sm);
    if (threadIdx.x == 0) partials[blockIdx.x] = blocksum;
}

// Kernel 3: deterministic final reduction of all block partials.
__global__ __launch_bounds__(256) void nce_finalize_kernel(
    const float* __restrict__ partials, float* __restrict__ out)
{
    __shared__ float sm[256];
    float acc = 0.0f;
    for (int i = threadIdx.x; i < N_PART; i += 256) acc += partials[i];
    const float total = block_reduce_256(acc, sm);
    if (threadIdx.x == 0) out[0] = total;
}

extern "C" void kernel_launch(void* const* d_in, const int* in_sizes, int n_in,
                              void* d_out, int out_size, void* d_ws, size_t ws_size,
                              hipStream_t stream)
{
    (void)in_sizes; (void)n_in; (void)out_size; (void)ws_size;

    const float* embed    = (const float*)d_in[0];
    const float* bias     = (const float*)d_in[1];
    const float* freq     = (const float*)d_in[2];
    const int*   targets  = (const int*)  d_in[3];
    const int*   contexts = (const int*)  d_in[4];
    const int*   noises   = (const int*)  d_in[5];
    // d_in[6] = noise_count (==16, compile-time constant here)

    float* ws       = (float*)d_ws;
    float* s_buf    = ws;                  // BATCH_N floats
    float* partials = ws + BATCH_N;        // N_PART floats

    nce_dot_kernel<<<K1_BLOCKS, 256, 0, stream>>>(
        embed, bias, freq, targets, contexts, s_buf, partials);

    nce_noise_kernel<<<K2_BLOCKS, 256, 0, stream>>>(
        freq, noises, s_buf, partials + K1_BLOCKS);

    nce_finalize_kernel<<<1, 256, 0, stream>>>(partials, (float*)d_out);
}